// TurboTokenUnpermuter_73169062854665
// MI455X (gfx1250) — compile-verified
//
#include <hip/hip_runtime.h>
#include <hip/hip_bf16.h>

// MoE unpermute+merge: out[t] = sum_over_chosen_experts( tokens[p] * probs[t,e] )
// Bandwidth-bound (AI ~0.35 FLOP/B): floor ~384 MB / 23.3 TB/s ~= 16.5 us.
// Strategy: invert the scatter into a gather (write-once output, read-once input),
// stream with 128-bit NT loads/stores, software-pipeline 8 loads/lane in the hot
// path, global_prefetch_b8 in the generic path. WMMA inapplicable by roofline.

typedef float f4 __attribute__((ext_vector_type(4)));

#define KMAX 8
#define GBLK 256  // gather block: 8 wave32

__global__ void zero_counts_kernel(int* __restrict__ count, int T) {
  int i = blockIdx.x * blockDim.x + threadIdx.x;
  if (i < T) count[i] = 0;
}

// For each permuted position p: record p into the inverse index of its target
// token and gather its merge probability. P=16K threads -> negligible cost.
__global__ void build_inv_kernel(const int* __restrict__ sidx,
                                 const int* __restrict__ eidx,
                                 const float* __restrict__ probs,
                                 int* __restrict__ count,
                                 int* __restrict__ inv,
                                 float* __restrict__ pprob,
                                 int P, int E, int K) {
  int p = blockIdx.x * blockDim.x + threadIdx.x;
  if (p >= P) return;
  int t = sidx[p];
  int e = eidx[p];
  pprob[p] = probs[t * E + e];
  int slot = atomicAdd(&count[t], 1);
  if (slot < K) inv[(size_t)t * K + slot] = p;
}

// One block (8 wave32) per output row: every output element written exactly once,
// every input element read exactly once. Source positions sorted ascending so the
// FP accumulation order is deterministic even though build_inv's slots race.
__global__ __launch_bounds__(GBLK) void gather_rows_kernel(
    const float* __restrict__ tokens,
    const int* __restrict__ count,
    const int* __restrict__ inv,
    const float* __restrict__ pprob,
    float* __restrict__ out,
    int H, int K) {
  const int t = blockIdx.x;
  int cnt = count[t];
  if (cnt > K) cnt = K;
  if (cnt > KMAX) cnt = KMAX;

  int pl[KMAX];
  float wl[KMAX];
  for (int i = 0; i < cnt; ++i) pl[i] = inv[(size_t)t * K + i];
  // insertion sort (cnt <= 8) for deterministic accumulation order
  for (int a = 1; a < cnt; ++a) {
    int key = pl[a];
    int b = a - 1;
    while (b >= 0 && pl[b] > key) { pl[b + 1] = pl[b]; --b; }
    pl[b + 1] = key;
  }
  for (int i = 0; i < cnt; ++i) wl[i] = pprob[pl[i]];

  const int H4 = H >> 2;
  const f4* __restrict__ tok4 = (const f4*)tokens;
  f4* __restrict__ out4 = (f4*)out;
  const size_t rowBase = (size_t)t * (size_t)H4;

  if (cnt == 2) {
    const f4* __restrict__ r0 = tok4 + (size_t)pl[0] * H4;
    const f4* __restrict__ r1 = tok4 + (size_t)pl[1] * H4;
    const float w0 = wl[0], w1 = wl[1];

    if (H4 == 4 * GBLK) {
      // Hot shape (H=4096): fully unrolled + software pipelined.
      // 8 outstanding global_load_b128 th:NT per lane before any use.
      const int j = threadIdx.x;
      f4 a0 = __builtin_nontemporal_load(r0 + j);
      f4 a1 = __builtin_nontemporal_load(r0 + j + GBLK);
      f4 a2 = __builtin_nontemporal_load(r0 + j + 2 * GBLK);
      f4 a3 = __builtin_nontemporal_load(r0 + j + 3 * GBLK);
      f4 b0 = __builtin_nontemporal_load(r1 + j);
      f4 b1 = __builtin_nontemporal_load(r1 + j + GBLK);
      f4 b2 = __builtin_nontemporal_load(r1 + j + 2 * GBLK);
      f4 b3 = __builtin_nontemporal_load(r1 + j + 3 * GBLK);
      f4 c0 = a0 * w0 + b0 * w1;
      f4 c1 = a1 * w0 + b1 * w1;
      f4 c2 = a2 * w0 + b2 * w1;
      f4 c3 = a3 * w0 + b3 * w1;
      __builtin_nontemporal_store(c0, out4 + rowBase + j);
      __builtin_nontemporal_store(c1, out4 + rowBase + j + GBLK);
      __builtin_nontemporal_store(c2, out4 + rowBase + j + 2 * GBLK);
      __builtin_nontemporal_store(c3, out4 + rowBase + j + 3 * GBLK);
    } else {
      for (int j = threadIdx.x; j < H4; j += blockDim.x) {
        int jn = j + (int)blockDim.x;
        if (jn < H4) {  // gfx1250 global_prefetch_b8
          __builtin_prefetch((const void*)(r0 + jn), 0, 0);
          __builtin_prefetch((const void*)(r1 + jn), 0, 0);
        }
        f4 v0 = __builtin_nontemporal_load(r0 + j);
        f4 v1 = __builtin_nontemporal_load(r1 + j);
        f4 acc = v0 * w0 + v1 * w1;
        __builtin_nontemporal_store(acc, out4 + rowBase + j);
      }
    }
  } else {
    for (int j = threadIdx.x; j < H4; j += blockDim.x) {
      f4 acc = {0.f, 0.f, 0.f, 0.f};
      for (int i = 0; i < cnt; ++i) {
        f4 v = __builtin_nontemporal_load(tok4 + (size_t)pl[i] * H4 + j);
        acc += v * wl[i];
      }
      __builtin_nontemporal_store(acc, out4 + rowBase + j);
    }
  }

  // scalar tail if H is not a multiple of 4 (not hit for H=4096)
  if (H & 3) {
    const int base = H4 << 2;
    for (int h = base + threadIdx.x; h < H; h += blockDim.x) {
      float acc = 0.f;
      for (int i = 0; i < cnt; ++i)
        acc += tokens[(size_t)pl[i] * H + h] * wl[i];
      out[(size_t)t * H + h] = acc;
    }
  }
}

extern "C" void kernel_launch(void* const* d_in, const int* in_sizes, int n_in,
                              void* d_out, int out_size, void* d_ws, size_t ws_size,
                              hipStream_t stream) {
  const float* tokens = (const float*)d_in[0];  // [P, H] f32
  const int*   sidx   = (const int*)d_in[1];    // [P]
  const int*   eidx   = (const int*)d_in[2];    // [P]
  const float* probs  = (const float*)d_in[3];  // [T, E] f32
  // d_in[4] = num_tokens device scalar: not readable during graph capture;
  // derive all shapes on host from sizes instead.

  const int P = in_sizes[1];
  const int H = (int)((long long)in_sizes[0] / P);
  const int T = (int)((long long)out_size / H);
  const int E = (int)((long long)in_sizes[3] / T);
  int K = P / T;
  if (K < 1) K = 1;
  if (K > KMAX) K = KMAX;

  // workspace layout: count[T] | inv[T*K] | pprob[P]  (~160 KB for this shape)
  int*   count = (int*)d_ws;
  int*   inv   = count + T;
  float* pprob = (float*)(inv + (size_t)T * K);

  zero_counts_kernel<<<(T + 255) / 256, 256, 0, stream>>>(count, T);
  build_inv_kernel<<<(P + 255) / 256, 256, 0, stream>>>(sidx, eidx, probs,
                                                        count, inv, pprob, P, E, K);
  gather_rows_kernel<<<T, GBLK, 0, stream>>>(tokens, count, inv, pprob,
                                             (float*)d_out, H, K);
}